// MetaplasticitySynapse_16063177687624
// MI455X (gfx1250) — compile-verified
//
#include <hip/hip_runtime.h>
#include <hip/hip_bf16.h>
#include <stddef.h>

typedef float v2f __attribute__((ext_vector_type(2)));
typedef float v8f __attribute__((ext_vector_type(8)));

#define P 4096
#define WIN 100
#define KCHUNK (P / 8)          /* 512 K-values per wave */

// exp(-DT/TAU_X) = exp(-0.01), exp(-DT/TAU_PLUS) = exp(-0.05)
#define DECAY_X   0.99004983f
#define DECAY_TR  0.95122945f

// ---------------------------------------------------------------------------
// GEMV via f32 WMMA: out[j] = sum_k pre[k] * W[k][j]
// One block (8 waves) per 16-column tile. Each wave consumes a 512-wide K
// chunk with V_WMMA_F32_16X16X4_F32 (A = pre broadcast across the 16 rows,
// B = W tile, all D rows identical). Wave partials reduced through LDS.
// 256 blocks x 8 waves = 2048 waves -> enough latency hiding for the
// 64 MB (L2-resident) weight sweep.
// ---------------------------------------------------------------------------
__global__ void __launch_bounds__(256)
gemv_wmma_kernel(const float* __restrict__ pre,
                 const float* __restrict__ W,
                 float* __restrict__ out_current) {
    __shared__ float spre[P];
    __shared__ float sred[8 * 16];

    const int tid = threadIdx.x;                 // 0..255 (8 waves)
    for (int i = tid; i < P; i += 256) spre[i] = pre[i];
    __syncthreads();

    const int lane = tid & 31;
    const int wv   = tid >> 5;                   // K-chunk id 0..7
    const int j0   = blockIdx.x << 4;            // column base (tile)
    const int n    = lane & 15;                  // N within tile
    const int hi   = lane >> 4;                  // K half (0: K0/K1, 1: K2/K3)

    const int kbeg = wv * KCHUNK;
    // 32-bit element offset from uniform base -> saddr + voffset loads,
    // row-pair distance (P floats = 16 KB) folds into the i24 immediate.
    unsigned off = (unsigned)((kbeg + hi * 2) * P + j0 + n);

    v8f c = {};
#pragma unroll 8
    for (int it = 0; it < KCHUNK; it += 4) {
        const int kk = kbeg + it + hi * 2;       // even -> 8B aligned
        v2f a = *reinterpret_cast<const v2f*>(spre + kk);   // ds_load_b64
        v2f b;
        b.x = W[off];
        b.y = W[off + P];
        // 8 args: (neg_a, A, neg_b, B, c_mod, C, reuse_a, reuse_b)
        c = __builtin_amdgcn_wmma_f32_16x16x4_f32(
                false, a, false, b, (short)0, c, false, false);
        off += 4u * P;
    }

    if (lane < 16) sred[wv * 16 + n] = c[0];
    __syncthreads();

    if (tid < 16) {
        float s = 0.f;
#pragma unroll
        for (int cix = 0; cix < 8; ++cix) s += sred[cix * 16 + tid];
        out_current[j0 + tid] = s;
    }
}

// ---------------------------------------------------------------------------
// new_weights = clip(weights, 0, 1); modulated_changes = 0  (float4 wide)
// Pure bandwidth: 64 MB read + 128 MB write ~ 8 us at 23.3 TB/s.
// ---------------------------------------------------------------------------
__global__ void __launch_bounds__(256)
weights_kernel(const float* __restrict__ W,
               float* __restrict__ out_neww,
               float* __restrict__ out_mod) {
    const size_t n4 = ((size_t)P * (size_t)P) / 4;   // 4,194,304 float4s
    const size_t stride = (size_t)gridDim.x * 256;
    const float4 zero = make_float4(0.f, 0.f, 0.f, 0.f);
    for (size_t i = (size_t)blockIdx.x * 256 + threadIdx.x; i < n4; i += stride) {
        float4 w = reinterpret_cast<const float4*>(W)[i];
        w.x = fminf(fmaxf(w.x, 0.f), 1.f);
        w.y = fminf(fmaxf(w.y, 0.f), 1.f);
        w.z = fminf(fmaxf(w.z, 0.f), 1.f);
        w.w = fminf(fmaxf(w.w, 0.f), 1.f);
        reinterpret_cast<float4*>(out_neww)[i] = w;
        reinterpret_cast<float4*>(out_mod)[i]  = zero;
    }
}

// ---------------------------------------------------------------------------
// All the small [P] state updates.
// ---------------------------------------------------------------------------
__global__ void __launch_bounds__(256)
vec_kernel(const float* __restrict__ pre_spikes,
           const float* __restrict__ post_spikes,
           const float* __restrict__ theta0,
           const float* __restrict__ x0,
           const float* __restrict__ pre_trace0,
           const float* __restrict__ post_trace0,
           const float* __restrict__ hist0,
           float* __restrict__ out_theta,
           float* __restrict__ out_x,
           float* __restrict__ out_recent,
           float* __restrict__ out_pretr,
           float* __restrict__ out_posttr) {
    const int p = blockIdx.x * 256 + threadIdx.x;
    if (p >= P) return;

    const float pre  = pre_spikes[p];
    const float post = post_spikes[p];

    // metaplastic activity variable
    out_x[p] = x0[p] * DECAY_X + post;

    // ring buffer: slot 0 replaced by post, mean over 100
    float s = 0.f;
    const float* __restrict__ h = hist0 + (size_t)p * WIN;
#pragma unroll 4
    for (int w = 1; w < WIN; ++w) s += h[w];
    const float recent = (s + post) * (1.0f / WIN);
    out_recent[p] = recent;

    // sliding threshold: theta += (recent - target)/tau_theta * dt, clipped
    float th = theta0[p] + (recent - 0.1f) * (0.001f / 10.0f);
    th = fminf(fmaxf(th, 0.05f), 0.2f);
    out_theta[p] = th;

    // STDP traces
    out_pretr[p]  = pre_trace0[p]  * DECAY_TR + pre;
    out_posttr[p] = post_trace0[p] * DECAY_TR + post;
}

extern "C" void kernel_launch(void* const* d_in, const int* in_sizes, int n_in,
                              void* d_out, int out_size, void* d_ws, size_t ws_size,
                              hipStream_t stream) {
    const float* pre_spikes  = (const float*)d_in[0];
    const float* post_spikes = (const float*)d_in[1];
    const float* weights     = (const float*)d_in[2];
    const float* theta0      = (const float*)d_in[3];
    const float* x0          = (const float*)d_in[4];
    const float* pre_trace0  = (const float*)d_in[5];
    const float* post_trace0 = (const float*)d_in[6];
    const float* hist0       = (const float*)d_in[7];

    float* out = (float*)d_out;
    const size_t PP = (size_t)P * (size_t)P;
    float* out_current = out;                         // [1,P]
    float* out_neww    = out + P;                     // [P,P]
    float* out_mod     = out + P + PP;                // [P,P] (zeros)
    float* out_theta   = out + P + 2 * PP;            // [1,P]
    float* out_x       = out_theta + P;               // [1,P]
    float* out_recent  = out_x + P;                   // [1,P]
    float* out_pretr   = out_recent + P;              // [1,P]
    float* out_posttr  = out_pretr + P;               // [1,P]

    // small vector updates
    vec_kernel<<<P / 256, 256, 0, stream>>>(
        pre_spikes, post_spikes, theta0, x0, pre_trace0, post_trace0, hist0,
        out_theta, out_x, out_recent, out_pretr, out_posttr);

    // clip + zeros over the two [P,P] tensors; also warms L2 with weights
    weights_kernel<<<2048, 256, 0, stream>>>(weights, out_neww, out_mod);

    // WMMA GEMV: one block (8 K-chunk waves) per 16-column tile
    gemv_wmma_kernel<<<P / 16, 256, 0, stream>>>(pre_spikes, weights, out_current);
}